// NRBS_1116691497544
// MI455X (gfx1250) — compile-verified
//
#include <hip/hip_runtime.h>
#include <math.h>

// Problem constants (match reference)
#define NTOT 16384
#define NLAT 20
#define MCL  10
#define KNB  32
#define BB   8
#define MUF  5.0f

typedef __attribute__((ext_vector_type(2))) float v2f;
typedef __attribute__((ext_vector_type(8))) float v8f;

// ---- workspace layout (floats) ----
#define NPART   128
#define KCHUNK  (NTOT / NPART)              // 128 -> 16 steps of K=8 (2x K=4 WMMA)
#define WS_PARTIALS 0                        // NPART * 2 tiles * 256
#define WS_ENCODED  (NPART * 512)            // 65536
#define WS_INVD     (WS_ENCODED + BB * NLAT) // +160
#define WS_EBL      (WS_INVD + MCL * NLAT * BB) // +1600  (total ~264KB)

// -------------------------------------------------------------------------
// Stage 1: encoded partial GEMM via V_WMMA_F32_16X16X4_F32.
// Branch-free padding: invalid rows are clamped to a legal address and the
// loaded values are multiplied by a per-lane 0/1 mask computed once, so EXEC
// never changes inside the loop (WMMA requires EXEC all-1s; no saveexec).
// Two independent accumulators per tile break the WMMA->WMMA RAW chain.
// A layout (16x4 f32): lane t: M=t&15, VGPR0 = K=(t<16?0:2), VGPR1 = K+1.
// B layout (4x16 f32): lane t: N=t&15, same K-half mapping.
// -------------------------------------------------------------------------
__global__ __launch_bounds__(32)
void encode_partial_wmma(const float* __restrict__ x,
                         const float* __restrict__ enc_w,
                         float* __restrict__ partials) {
  const int t = threadIdx.x;
  const int p = blockIdx.x;
  const int M = t & 15;
  const int Nn = t & 15;
  const int half = (t >> 4) << 1;                 // 0 (lanes 0-15) or 2
  const float maskA  = (M < BB) ? 1.f : 0.f;      // pad rows b=8..15 -> 0
  const float maskB1 = (Nn < NLAT - 16) ? 1.f : 0.f; // pad cols l=20..31 -> 0
  const int rowA  = (M < BB) ? M : 0;             // clamped, always legal
  const int rowB1 = (Nn < NLAT - 16) ? (16 + Nn) : 16;
  const float* pa  = x     + (size_t)rowA  * NTOT + half;
  const float* pb0 = enc_w + (size_t)Nn    * NTOT + half;  // l = 0..15 valid
  const float* pb1 = enc_w + (size_t)rowB1 * NTOT + half;  // l = 16..19
  v8f acc0a = {}, acc0b = {}, acc1a = {}, acc1b = {};
  const int n0base = p * KCHUNK;
  for (int s = 0; s < KCHUNK; s += 8) {
    const int n0 = n0base + s;
    const float2 av0 = *(const float2*)(pa  + n0);
    const float2 bv0 = *(const float2*)(pb0 + n0);
    const float2 cv0 = *(const float2*)(pb1 + n0);
    const float2 av1 = *(const float2*)(pa  + n0 + 4);
    const float2 bv1 = *(const float2*)(pb0 + n0 + 4);
    const float2 cv1 = *(const float2*)(pb1 + n0 + 4);
    v2f a0; a0.x = av0.x * maskA;  a0.y = av0.y * maskA;
    v2f b0; b0.x = bv0.x;          b0.y = bv0.y;
    v2f c0; c0.x = cv0.x * maskB1; c0.y = cv0.y * maskB1;
    v2f a1; a1.x = av1.x * maskA;  a1.y = av1.y * maskA;
    v2f b1; b1.x = bv1.x;          b1.y = bv1.y;
    v2f c1; c1.x = cv1.x * maskB1; c1.y = cv1.y * maskB1;
    acc0a = __builtin_amdgcn_wmma_f32_16x16x4_f32(false, a0, false, b0,
                                                  (short)0, acc0a, false, false);
    acc1a = __builtin_amdgcn_wmma_f32_16x16x4_f32(false, a0, false, c0,
                                                  (short)0, acc1a, false, false);
    acc0b = __builtin_amdgcn_wmma_f32_16x16x4_f32(false, a1, false, b1,
                                                  (short)0, acc0b, false, false);
    acc1b = __builtin_amdgcn_wmma_f32_16x16x4_f32(false, a1, false, c1,
                                                  (short)0, acc1b, false, false);
  }
  // C/D layout: VGPR i, lane t: M = i + (t<16?0:8), N = t&15.
  float* out0 = partials + (size_t)(p * 2 + 0) * 256;
  float* out1 = partials + (size_t)(p * 2 + 1) * 256;
#pragma unroll
  for (int i = 0; i < 8; ++i) {
    out0[i * 32 + t] = acc0a[i] + acc0b[i];
    out1[i * 32 + t] = acc1a[i] + acc1b[i];
  }
}

// -------------------------------------------------------------------------
// Stage 1b: reduce partials over NPART blocks, add bias -> encoded[b][l]
// For b<8: value(b, N) sits at VGPR i=b, lane t=N (low half).
// -------------------------------------------------------------------------
__global__ __launch_bounds__(256)
void encode_reduce(const float* __restrict__ partials,
                   const float* __restrict__ enc_b,
                   float* __restrict__ encoded) {
  const int tid = threadIdx.x;
  if (tid >= BB * NLAT) return;
  const int b = tid / NLAT, l = tid % NLAT;
  const int tile = l >> 4, Nn = l & 15;
  float s = 0.f;
  for (int p = 0; p < NPART; ++p)
    s += partials[(size_t)(p * 2 + tile) * 256 + b * 32 + Nn];
  encoded[b * NLAT + l] = s + enc_b[l];
}

// -------------------------------------------------------------------------
// Stage 2: bandwidth MLP. Only 1600 distinct inverse-denominators exist
// (B x NLAT x MCL); store as invd_t[m][l][b] for coalesced per-lane reads.
// Also transpose encoded -> ebl[l][b].
// -------------------------------------------------------------------------
__global__ __launch_bounds__(256)
void bandwidth_kernel(const float* __restrict__ encoded,
                      const float* __restrict__ bw_w,
                      const float* __restrict__ bw_b,
                      float* __restrict__ invd_t,
                      float* __restrict__ ebl) {
  const int tid = blockIdx.x * blockDim.x + threadIdx.x;
  if (tid < BB * NLAT * MCL) {
    const int b = tid / (NLAT * MCL);
    const int j = tid % (NLAT * MCL);
    float acc = bw_b[j];
    for (int l2 = 0; l2 < NLAT; ++l2)
      acc = fmaf(encoded[b * NLAT + l2], bw_w[j * NLAT + l2], acc);
    const float s = 1.0f / (1.0f + expf(-acc));
    const float lo = 2.0f / 60.0f / MUF;
    const float hi = 1.0f / 60.0f;
    const float bw = (hi - lo) * s + lo;
    const int l = j / MCL, m = j % MCL;
    const float d = bw * MUF;
    invd_t[((size_t)m * NLAT + l) * BB + b] = 1.0f / (d * d);
  }
  if (tid < BB * NLAT) {
    const int l = tid / BB, b = tid % BB;
    ebl[tid] = encoded[b * NLAT + l];
  }
}

// -------------------------------------------------------------------------
// Stage 3: windowed neighbor convolution + final contraction (dominant cost:
// ~84M f32 VALU iterations over L2-resident data -> VALU-bound).
// One wave per point n (K=32 == wave32). Lane remap (b = t&7, kc = t>>3):
// each lane accumulates 8 k's in-register; g/d2 broadcast via ds_bpermute
// (__shfl); normalization reduction is only 2 shfl_xor stages (4 lanes).
// -------------------------------------------------------------------------
#define WAVES_TOTAL 512
#define NPW (NTOT / WAVES_TOTAL)   // 32 points per wave

__global__ __launch_bounds__(256)
void conv_kernel(const float* __restrict__ ndist,
                 const int* __restrict__ nbid,
                 const int* __restrict__ labels,
                 const float* __restrict__ dec_w,
                 const float* __restrict__ invd_t,
                 const float* __restrict__ ebl,
                 float* __restrict__ out) {
  const int t  = threadIdx.x & 31;
  const int wg = (blockIdx.x * blockDim.x + threadIdx.x) >> 5; // 0..511
  const int b  = t & 7;
  const int kc = t >> 3;
  for (int it = 0; it < NPW; ++it) {
    const int n = wg * NPW + it;
    if (it + 1 < NPW) {
      __builtin_prefetch(ndist + (size_t)(n + 1) * KNB, 0, 1);
      __builtin_prefetch(nbid  + (size_t)(n + 1) * KNB, 0, 1);
    }
    const float nd  = ndist[(size_t)n * KNB + t];   // lane = k (coalesced)
    const int   nid = nbid [(size_t)n * KNB + t];
    const float d2  = nd * nd;
    const int   m   = labels[n];                    // wave-uniform -> scalar
    float d2v[8];
#pragma unroll
    for (int j = 0; j < 8; ++j)
      d2v[j] = __shfl(d2, kc * 8 + j, 32);
    float outacc = 0.f;
    const float* invrow = invd_t + (size_t)m * NLAT * BB;
    for (int l = 0; l < NLAT; ++l) {
      const float g   = dec_w[(size_t)l * NTOT + nid]; // one gather/lane (L2-resident)
      const float inv = invrow[l * BB + b];
      float wsum = 0.f, ps = 0.f;
#pragma unroll
      for (int j = 0; j < 8; ++j) {
        const float gv = __shfl(g, kc * 8 + j, 32);
        const float w  = fmaxf(0.f, fmaf(-d2v[j], inv, 1.0f));
        wsum += w;
        ps = fmaf(gv, w, ps);
      }
      // combine the 4 k-chunks holding the same b (xor lanes 8, 16)
      wsum += __shfl_xor(wsum, 8, 32);
      wsum += __shfl_xor(wsum, 16, 32);
      ps   += __shfl_xor(ps, 8, 32);
      ps   += __shfl_xor(ps, 16, 32);
      const float conv = ps / wsum;
      outacc = fmaf(ebl[l * BB + b], conv, outacc);
    }
    if (t < 8)
      out[(size_t)b * NTOT + n] = outacc;
  }
}

// -------------------------------------------------------------------------
extern "C" void kernel_launch(void* const* d_in, const int* in_sizes, int n_in,
                              void* d_out, int out_size, void* d_ws, size_t ws_size,
                              hipStream_t stream) {
  const float* x      = (const float*)d_in[0];
  const float* enc_w  = (const float*)d_in[1];
  const float* enc_b  = (const float*)d_in[2];
  const float* dec_w  = (const float*)d_in[3];
  const float* bw_w   = (const float*)d_in[4];
  const float* bw_b   = (const float*)d_in[5];
  const float* ndist  = (const float*)d_in[6];
  const int*   nbid   = (const int*)d_in[7];
  const int*   labels = (const int*)d_in[8];
  float* out = (float*)d_out;

  float* ws       = (float*)d_ws;
  float* partials = ws + WS_PARTIALS;
  float* encoded  = ws + WS_ENCODED;
  float* invd_t   = ws + WS_INVD;
  float* ebl      = ws + WS_EBL;

  encode_partial_wmma<<<NPART, 32, 0, stream>>>(x, enc_w, partials);
  encode_reduce<<<1, 256, 0, stream>>>(partials, enc_b, encoded);
  bandwidth_kernel<<<7, 256, 0, stream>>>(encoded, bw_w, bw_b, invd_t, ebl);
  conv_kernel<<<64, 256, 0, stream>>>(ndist, nbid, labels, dec_w, invd_t, ebl, out);
}